// MultiFrequencyResponseModule_48868137893967
// MI455X (gfx1250) — compile-verified
//
#include <hip/hip_runtime.h>
#include <cstdint>

typedef __attribute__((ext_vector_type(2))) float v2f;
typedef __attribute__((ext_vector_type(8))) float v8f;

#define BB 8
#define CC 128
#define HH0 256
#define WW0 256

// ---- CDNA5 async global->LDS helpers (gfx1250) -----------------------------
__device__ __forceinline__ uint32_t lds_off_of(const void* p) {
  // Flat LDS-aperture addresses carry the LDS byte offset in the low 32 bits.
  return (uint32_t)(uintptr_t)p;
}
__device__ __forceinline__ void async_b128(uint32_t lds, uint64_t gbase, uint32_t goff) {
  asm volatile("global_load_async_to_lds_b128 %0, %1, %2"
               :: "v"(lds), "v"(goff), "s"(gbase) : "memory");
}
__device__ __forceinline__ void async_b64(uint32_t lds, uint64_t gbase, uint32_t goff) {
  asm volatile("global_load_async_to_lds_b64 %0, %1, %2"
               :: "v"(lds), "v"(goff), "s"(gbase) : "memory");
}
__device__ __forceinline__ void wait_async0() {
  asm volatile("s_wait_asynccnt 0" ::: "memory");
}

// ---- Fused Haar-WT (WMMA butterflies) + depthwise 5x5 on subbands ----------
// One workgroup: one (b, c, 32x32 subband tile). Input tile 72x72 staged in LDS.
__global__ __launch_bounds__(256) void k_wt_dw(
    const float* __restrict__ in, float* __restrict__ tag, float* __restrict__ ll,
    const float* __restrict__ wtf, const float* __restrict__ wconv,
    const float* __restrict__ wsc, int Hi, int Wi, int tilesX, int tilesY, int writeLL)
{
  const int hh = Hi >> 1, ww = Wi >> 1;
  __shared__ float tin[72 * 96];      // input tile, stride 96 (bank-friendly)
  __shared__ float sub[4][36 * 37];   // 4 subband planes, 36x36 (+2 halo each side)

  int bid = blockIdx.x;
  const int tX = bid % tilesX; bid /= tilesX;
  const int tY = bid % tilesY; bid /= tilesY;
  const int c = bid % CC;
  const int b = bid / CC;

  const int tid = threadIdx.x;
  const int oy0 = tY * 32, ox0 = tX * 32;      // subband-tile origin
  const int iy0 = 2 * oy0 - 4, ix0 = 2 * ox0 - 4;
  const size_t inBase = (size_t)(b * CC + c) * Hi * Wi;

  const bool interior = (iy0 >= 0) && (ix0 >= 0) && (iy0 + 72 <= Hi) && (ix0 + 72 <= Wi);
  if (interior) {
    const uint64_t gb = (uint64_t)(uintptr_t)(in + inBase);
    const uint32_t tb = lds_off_of(&tin[0]);
    for (int e = tid; e < 72 * 18; e += 256) {   // 18 x b128 per 72-float row
      const int r = e / 18, s = e - r * 18;
      async_b128(tb + (uint32_t)((r * 96 + s * 4) * 4), gb,
                 (uint32_t)(((iy0 + r) * Wi + ix0 + s * 4) * 4));
    }
    wait_async0();
  } else {
    for (int e = tid; e < 72 * 72; e += 256) {
      const int r = e / 72, q = e - r * 72;
      const int iy = iy0 + r, ix = ix0 + q;
      float v = 0.f;
      if (iy >= 0 && iy < Hi && ix >= 0 && ix < Wi) v = in[inBase + (size_t)iy * Wi + ix];
      tin[r * 96 + q] = v;
    }
  }
  __syncthreads();

  // Haar butterflies on the matrix pipe: D(16x16) = H(16x4) x Blocks(4x16).
  // Every lane holds s0..s3 at constant components (A rows replicate mod 4):
  // lanes 0-15 in d[0..3] (M=0..3), lanes 16-31 in d[0..3] (M=8..11 == 0..3).
  const int lane = tid & 31, wv = tid >> 5;
  const int hf = lane >> 4, n16 = lane & 15, m4 = n16 & 3;
  v2f aop;                                   // A 16x4: lanes0-15 K=0/1, lanes16-31 K=2/3
  aop.x = wtf[m4 * 4 + hf * 2 + 0];
  aop.y = wtf[m4 * 4 + hf * 2 + 1];
  for (int row = wv; row < 36; row += 8) {   // rows across 8 waves, no div/mod
    const float* trow = &tin[(2 * row + hf) * 96];
    float* s0p = &sub[hf * 2 + 0][row * 37];
    float* s1p = &sub[hf * 2 + 1][row * 37];
#pragma unroll
    for (int ch = 0; ch < 3; ++ch) {         // 3 chunks of 16 blocks (16,16,4 valid)
      const int sxL = ch * 16 + n16;
      v2f bop; bop.x = trow[2 * sxL]; bop.y = trow[2 * sxL + 1];
      v8f cz = {};
      v8f d = __builtin_amdgcn_wmma_f32_16x16x4_f32(false, aop, false, bop,
                                                    (short)0, cz, false, false);
      const float e0 = d[0], e1 = d[1], e2 = d[2], e3 = d[3];
      const float vlo = hf ? e2 : e0;        // plane hf*2
      const float vhi = hf ? e3 : e1;        // plane hf*2+1
      if (sxL < 36) { s0p[sxL] = vlo; s1p[sxL] = vhi; }
    }
  }
  __syncthreads();

  // Depthwise 5x5 (SAME) per subband channel; weights via uniform scalar loads.
  const int px = tid & 31, py0 = tid >> 5;
  const float* wc = wconv + (size_t)(4 * c) * 25;
  for (int rr = 0; rr < 4; ++rr) {
    const int py = py0 + rr * 8;
#pragma unroll
    for (int s = 0; s < 4; ++s) {
      float acc = 0.f;
#pragma unroll
      for (int dy = 0; dy < 5; ++dy)
#pragma unroll
        for (int dx = 0; dx < 5; ++dx)
          acc += wc[s * 25 + dy * 5 + dx] * sub[s][(py + dy) * 37 + (px + dx)];
      acc *= wsc[4 * c + s];
      tag[((size_t)(b * 4 * CC + 4 * c + s) * hh + (oy0 + py)) * (size_t)ww + (ox0 + px)] = acc;
    }
    if (writeLL)
      ll[((size_t)(b * CC + c) * hh + (oy0 + py)) * (size_t)ww + (ox0 + px)] =
          sub[0][(py + 2) * 37 + (px + 2)];
  }
}

// ---- Base depthwise 5x5 conv + bias + scale --------------------------------
__global__ __launch_bounds__(256) void k_base(
    const float* __restrict__ x, float* __restrict__ out,
    const float* __restrict__ bw, const float* __restrict__ bbias,
    const float* __restrict__ bscale)
{
  __shared__ float t[36 * 40];
  int bid = blockIdx.x;
  const int tX = bid & 7; bid >>= 3;
  const int tY = bid & 7; bid >>= 3;
  const int c = bid % CC;
  const int b = bid / CC;
  const int tid = threadIdx.x;
  const int oy0 = tY * 32, ox0 = tX * 32;
  const int iy0 = oy0 - 2, ix0 = ox0 - 2;
  const size_t base = (size_t)(b * CC + c) * HH0 * WW0;

  const bool interior = (iy0 >= 0) && (ix0 >= 0) && (iy0 + 36 <= HH0) && (ix0 + 36 <= WW0);
  if (interior) {
    const uint64_t gb = (uint64_t)(uintptr_t)(x + base);
    const uint32_t tb = lds_off_of(&t[0]);
    for (int e = tid; e < 36 * 18; e += 256) {  // 18 x b64 per 36-float row
      const int r = e / 18, s = e - r * 18;
      async_b64(tb + (uint32_t)((r * 40 + s * 2) * 4), gb,
                (uint32_t)(((iy0 + r) * WW0 + ix0 + s * 2) * 4));
    }
    wait_async0();
  } else {
    for (int e = tid; e < 36 * 36; e += 256) {
      const int r = e / 36, q = e - r * 36;
      const int iy = iy0 + r, ix = ix0 + q;
      float v = 0.f;
      if (iy >= 0 && iy < HH0 && ix >= 0 && ix < WW0) v = x[base + (size_t)iy * WW0 + ix];
      t[r * 40 + q] = v;
    }
  }
  __syncthreads();

  const int px = tid & 31, py0 = tid >> 5;
  const float* w = bw + c * 25;
  const float bias = bbias[c], scale = bscale[c];
  for (int rr = 0; rr < 4; ++rr) {
    const int py = py0 + rr * 8;
    float acc = 0.f;
#pragma unroll
    for (int dy = 0; dy < 5; ++dy)
#pragma unroll
      for (int dx = 0; dx < 5; ++dx)
        acc += w[dy * 5 + dx] * t[(py + dy) * 40 + (px + dx)];
    out[base + (size_t)(oy0 + py) * WW0 + (ox0 + px)] = scale * (acc + bias);
  }
}

// ---- Inverse Haar via WMMA: out 2x2 block = H x [ll+nll, h1, h2, h3] -------
// One wave handles 16 consecutive subband positions per WMMA; level 0
// accumulates onto the base result already in d_out.
__global__ __launch_bounds__(256) void k_iwt(
    const float* __restrict__ tag, const float* __restrict__ nll,
    float* __restrict__ outp, const float* __restrict__ iwtf,
    int hh, int ww, int accumulate)
{
  const int tid = threadIdx.x;
  const int lane = tid & 31, wv = tid >> 5;
  const int hf = lane >> 4, n16 = lane & 15, m4 = n16 & 3;
  const int cw = ww >> 4;
  long g = (long)blockIdx.x * 8 + wv;
  const int sxc = (int)(g % cw); g /= cw;
  const int sy = (int)(g % hh); g /= hh;
  const int c = (int)(g % CC);
  const int b = (int)(g / CC);
  const int sx = sxc * 16 + n16;
  const size_t hw = (size_t)hh * ww;

  const size_t tb = ((size_t)(b * 4 * CC + 4 * c + hf * 2)) * hw + (size_t)sy * ww + sx;
  v2f bop;
  bop.x = tag[tb];            // lanes0-15: s0 ; lanes16-31: s2
  bop.y = tag[tb + hw];       // lanes0-15: s1 ; lanes16-31: s3
  if (hf == 0 && nll)
    bop.x += nll[(size_t)(b * CC + c) * hw + (size_t)sy * ww + sx];

  v2f aop;                    // A[m][s] = rec_s[r,q], m=2r+q, rows replicate mod 4
  aop.x = iwtf[(hf * 2 + 0) * 4 + m4];
  aop.y = iwtf[(hf * 2 + 1) * 4 + m4];
  v8f cz = {};
  v8f d = __builtin_amdgcn_wmma_f32_16x16x4_f32(false, aop, false, bop,
                                                (short)0, cz, false, false);
  // lanes0-15 write the top pixel row (d[0],d[1]); lanes16-31 the bottom (d[2],d[3]).
  const float e0 = d[0], e1 = d[1], e2 = d[2], e3 = d[3];
  float r0 = hf ? e2 : e0;
  float r1 = hf ? e3 : e1;
  const int W2 = ww * 2;
  const size_t o = ((size_t)(b * CC + c) * (hh * 2) + (2 * sy + hf)) * (size_t)W2
                   + 2 * (size_t)sx;
  if (accumulate) { r0 += outp[o]; r1 += outp[o + 1]; }
  float2 st; st.x = r0; st.y = r1;
  *(float2*)&outp[o] = st;
}

// ---- Host launch -----------------------------------------------------------
extern "C" void kernel_launch(void* const* d_in, const int* in_sizes, int n_in,
                              void* d_out, int out_size, void* d_ws, size_t ws_size,
                              hipStream_t stream) {
  (void)in_sizes; (void)n_in; (void)out_size; (void)ws_size;
  const float* x    = (const float*)d_in[0];
  const float* wtf  = (const float*)d_in[1];
  const float* iwtf = (const float*)d_in[2];
  const float* bw   = (const float*)d_in[3];
  const float* bb   = (const float*)d_in[4];
  const float* bs   = (const float*)d_in[5];
  const float* wcw  = (const float*)d_in[6];
  const float* wsc  = (const float*)d_in[7];
  float* out = (float*)d_out;
  float* ws  = (float*)d_ws;

  // Workspace layout (floats); nll buffers alias dead LL buffers. ~436 MB.
  float* tag0 = ws;
  float* tag1 = tag0 + (size_t)BB * 4 * CC * 128 * 128;
  float* tag2 = tag1 + (size_t)BB * 4 * CC * 64 * 64;
  float* cur1 = tag2 + (size_t)BB * 4 * CC * 32 * 32;
  float* cur2 = cur1 + (size_t)BB * CC * 128 * 128;
  float* nll2 = cur2;   // cur2 dead after level-2 WT
  float* nll1 = cur1;   // cur1 dead after level-1 WT

  const dim3 blk(256);
  const size_t w4c25 = (size_t)4 * CC * 25;

  // Forward pyramid (fused WT + depthwise 5x5 per level)
  k_wt_dw<<<dim3(BB * CC * 4 * 4), blk, 0, stream>>>(x,    tag0, cur1, wtf,
      wcw + 0 * w4c25, wsc + 0 * 4 * CC, 256, 256, 4, 4, 1);
  k_wt_dw<<<dim3(BB * CC * 2 * 2), blk, 0, stream>>>(cur1, tag1, cur2, wtf,
      wcw + 1 * w4c25, wsc + 1 * 4 * CC, 128, 128, 2, 2, 1);
  k_wt_dw<<<dim3(BB * CC * 1 * 1), blk, 0, stream>>>(cur2, tag2, nullptr, wtf,
      wcw + 2 * w4c25, wsc + 2 * 4 * CC, 64, 64, 1, 1, 0);

  // Base path writes d_out fully (d_out is poisoned by harness).
  k_base<<<dim3(BB * CC * 64), blk, 0, stream>>>(x, out, bw, bb, bs);

  // Inverse chain; level 0 accumulates onto base in d_out.
  k_iwt<<<dim3(BB * CC * 32 * (32 / 16) / 8), blk, 0, stream>>>(tag2, nullptr, nll2, iwtf, 32, 32, 0);
  k_iwt<<<dim3(BB * CC * 64 * (64 / 16) / 8), blk, 0, stream>>>(tag1, nll2,    nll1, iwtf, 64, 64, 0);
  k_iwt<<<dim3(BB * CC * 128 * (128 / 16) / 8), blk, 0, stream>>>(tag0, nll1,  out,  iwtf, 128, 128, 1);
}